// CenterLoss_8297876815993
// MI455X (gfx1250) — compile-verified
//
#include <hip/hip_runtime.h>
#include <hip/hip_bf16.h>

// CenterLoss, but only the labeled column of the distance matrix is needed:
//   dist[b] = |x[b]|^2 + |c_lb|^2 - 2*<x[b], c_lb>;  loss = mean(clip(dist,1e-12,1e12))
// Memory-bound: ~32 MB total traffic -> ~1.4 us at 23.3 TB/s. The dot product is
// computed with V_WMMA_F32_16X16X4_F32 (full fp32, matching the reference GEMM
// contraction); we keep the 16x16 tile's diagonal.

typedef __attribute__((ext_vector_type(2))) float v2f;
typedef __attribute__((ext_vector_type(4))) float v4f;
typedef __attribute__((ext_vector_type(8))) float v8f;

#define BATCH   2048
#define FEAT    2048
#define TILE_M  16
#define NWAVES  8
#define KSLICE  (FEAT / NWAVES)    // 256 K-values per wave
#define NTILES  (BATCH / TILE_M)   // 128 blocks

__global__ __launch_bounds__(NWAVES * 32)
void center_loss_main(const float* __restrict__ x,
                      const int*   __restrict__ labels,
                      const float* __restrict__ centers,
                      float*       __restrict__ tile_partial)
{
    __shared__ float s_part[NWAVES][TILE_M];
    __shared__ float s_dist[TILE_M];

    const int tile = blockIdx.x;          // 16-row tile of the batch
    const int lane = threadIdx.x & 31;
    const int wave = threadIdx.x >> 5;    // 0..7, owns a 256-wide K slice
    const int m    = lane & 15;           // row within tile (M for A, N for B)
    const int half = lane >> 4;           // K sub-slot selector
    const int row  = tile * TILE_M + m;

    const int lbl = labels[row];          // gather: one center row per batch row
    const float* __restrict__ xr = x       + (size_t)row * FEAT;
    const float* __restrict__ cr = centers + (size_t)lbl * FEAT;

    // Lane covers K = [kb + k, kb + k + 3] each iteration; lanes L and L+16
    // cover complementary 4-wide halves of an 8-wide chunk.
    const int kb = wave * KSLICE + half * 4;

    v8f  acc = {};                 // 16x16 fp32 tile; diagonal = partial dots
    float xsq = 0.0f, csq = 0.0f;  // per-lane partial squared norms

    #pragma unroll 4
    for (int k = 0; k < KSLICE; k += 8) {
        v4f fx = *(const v4f*)(xr + kb + k);   // global_load_b128
        v4f fc = *(const v4f*)(cr + kb + k);   // global_load_b128 (gathered)

        v2f a0 = { fx.x, fx.y };  v2f b0 = { fc.x, fc.y };
        v2f a1 = { fx.z, fx.w };  v2f b1 = { fc.z, fc.w };

        // D = A x B + C : A = x tile (16xK slice), B = gathered-centers^T.
        // A/B f32 layouts are lane-symmetric, so the same per-lane K
        // permutation is applied to both operands -> dot over K is exact.
        acc = __builtin_amdgcn_wmma_f32_16x16x4_f32(
                  false, a0, false, b0, (short)0, acc, false, false);
        acc = __builtin_amdgcn_wmma_f32_16x16x4_f32(
                  false, a1, false, b1, (short)0, acc, false, false);

        xsq = fmaf(fx.x, fx.x, fmaf(fx.y, fx.y,
              fmaf(fx.z, fx.z, fmaf(fx.w, fx.w, xsq))));
        csq = fmaf(fc.x, fc.x, fmaf(fc.y, fc.y,
              fmaf(fc.z, fc.z, fmaf(fc.w, fc.w, csq))));
    }

    // Lanes L and L+16 hold complementary K-halves of row (L%16): combine.
    xsq += __shfl_xor(xsq, 16, 32);
    csq += __shfl_xor(csq, 16, 32);

    // Diagonal ownership for the 16x16 f32 C/D layout:
    //   lanes 0-7  : D[m][m], m = lane,      vgpr = lane
    //   lanes 24-31: D[m][m], m = lane - 16, vgpr = lane - 24
    const bool ownLo = (lane < 8);
    const bool ownHi = (lane >= 24);
    if (ownLo || ownHi) {
        const int idx = ownLo ? lane : (lane - 24);
        float dotd = acc[0];
        #pragma unroll
        for (int r = 1; r < 8; ++r) if (idx == r) dotd = acc[r];
        // Partial (this wave's K slice) of |x|^2 + |c|^2 - 2<x,c>; linear in K,
        // so clamp happens only after summing all waves.
        s_part[wave][m] = xsq + csq - 2.0f * dotd;
    }
    __syncthreads();

    // Fixed-order cross-wave reduction (deterministic), then clamp per row.
    if (threadIdx.x < TILE_M) {
        float d = 0.0f;
        #pragma unroll
        for (int w = 0; w < NWAVES; ++w) d += s_part[w][threadIdx.x];
        d = fminf(fmaxf(d, 1e-12f), 1e12f);
        s_dist[threadIdx.x] = d;
    }
    __syncthreads();

    if (threadIdx.x == 0) {
        float s = 0.0f;
        #pragma unroll
        for (int i = 0; i < TILE_M; ++i) s += s_dist[i];
        tile_partial[tile] = s;
    }
}

// Fixed-order final reduction over 128 tile partials -> mean. Deterministic,
// trivial cost, graph-capture safe (no atomics, no host sync).
__global__ void center_loss_final(const float* __restrict__ tile_partial,
                                  float* __restrict__ out)
{
    if (threadIdx.x == 0 && blockIdx.x == 0) {
        float s = 0.0f;
        for (int i = 0; i < NTILES; ++i) s += tile_partial[i];
        out[0] = s * (1.0f / (float)BATCH);
    }
}

extern "C" void kernel_launch(void* const* d_in, const int* in_sizes, int n_in,
                              void* d_out, int out_size, void* d_ws, size_t ws_size,
                              hipStream_t stream) {
    const float* x       = (const float*)d_in[0];
    const int*   labels  = (const int*)  d_in[1];
    const float* centers = (const float*)d_in[2];
    float*       ws      = (float*)d_ws;          // NTILES floats of scratch

    center_loss_main<<<NTILES, NWAVES * 32, 0, stream>>>(x, labels, centers, ws);
    center_loss_final<<<1, 32, 0, stream>>>(ws, (float*)d_out);
}